// TransitionDownWithDistFea_65592740544740
// MI455X (gfx1250) — compile-verified
//
#include <hip/hip_runtime.h>
#include <hip/hip_bf16.h>

// ---------------- problem constants ----------------
constexpr int B_ = 2;
constexpr int N_ = 8192;
constexpr int S_ = 4096;
constexpr int GEN_CH = 28;   // 6 + 6 + 16
constexpr int IN_CH = 64;
constexpr int OUT_CH = 128;

typedef float v2f __attribute__((ext_vector_type(2)));
typedef float v8f __attribute__((ext_vector_type(8)));

__device__ __forceinline__ v8f wmma_f32_k4(v2f a, v2f b, v8f c) {
    // V_WMMA_F32_16X16X4_F32 : D = A(16x4) * B(4x16) + C(16x16), all f32
    return __builtin_amdgcn_wmma_f32_16x16x4_f32(
        /*neg_a=*/false, a, /*neg_b=*/false, b,
        /*c_mod=*/(short)0, c, /*reuse_a=*/false, /*reuse_b=*/false);
}

// async copy of one dword: global -> LDS, tracked by ASYNCcnt
__device__ __forceinline__ void async_dword(void* lds_ptr, const void* gptr) {
    unsigned lds_off = (unsigned)(unsigned long long)lds_ptr;   // low 32 bits = LDS byte offset
    unsigned long long ga = (unsigned long long)gptr;
    asm volatile("global_load_async_to_lds_b32 %0, %1, off"
                 :: "v"(lds_off), "v"(ga) : "memory");
}

// distance identical to reference formula: sqrt(max(sq_i + sq_j - 2*dot, 1e-12))
__device__ __forceinline__ float pdist(const float* __restrict__ xyz,
                                       const float* __restrict__ sq,
                                       int base, int i, int j) {
    const float* pi = xyz + (size_t)(base + i) * 3;
    const float* pj = xyz + (size_t)(base + j) * 3;
    float dot = pi[0] * pj[0] + pi[1] * pj[1] + pi[2] * pj[2];
    float d2 = sq[base + i] + sq[base + j] - 2.0f * dot;
    return sqrtf(fmaxf(d2, 1e-12f));
}

__device__ __forceinline__ void ins4(float d, int j,
                                     float& bd0, float& bd1, float& bd2, float& bd3,
                                     int& bi0, int& bi1, int& bi2, int& bi3) {
    if (d < bd3) {
        if (d < bd2) {
            bd3 = bd2; bi3 = bi2;
            if (d < bd1) {
                bd2 = bd1; bi2 = bi1;
                if (d < bd0) { bd1 = bd0; bi1 = bi0; bd0 = d; bi0 = j; }
                else         { bd1 = d;  bi1 = j; }
            } else { bd2 = d; bi2 = j; }
        } else { bd3 = d; bi3 = j; }
    }
}

// ---------------- tiny prep kernels ----------------
__global__ void sq_kernel(const float* __restrict__ xyz, float* __restrict__ sq) {
    int i = blockIdx.x * 256 + threadIdx.x;
    if (i < B_ * N_) {
        const float* p = xyz + (size_t)i * 3;
        sq[i] = p[0] * p[0] + p[1] * p[1] + p[2] * p[2];
    }
}

__global__ void sampled_xyz_kernel(const float* __restrict__ xyz,
                                   const int* __restrict__ sidx,
                                   float* __restrict__ out) {
    int i = blockIdx.x * 256 + threadIdx.x;   // i = b*S + s
    if (i < B_ * S_) {
        int b = i >> 12;                      // / S_
        int g = sidx[i];
        const float* p = xyz + ((size_t)b * N_ + g) * 3;
        out[(size_t)i * 3 + 0] = p[0];
        out[(size_t)i * 3 + 1] = p[1];
        out[(size_t)i * 3 + 2] = p[2];
    }
}

// ---------------- KNN over all N points (queries = sampled) ----------------
__global__ __launch_bounds__(256) void knn1_kernel(const float* __restrict__ xyz,
                                                   const float* __restrict__ sq,
                                                   const int* __restrict__ sidx,
                                                   int* __restrict__ anchors,
                                                   float* __restrict__ trans) {
    __shared__ float cx[2][256], cy[2][256], cz[2][256], cs[2][256];
    int tid = threadIdx.x;
    int b = blockIdx.y;
    int s = blockIdx.x * 256 + tid;
    int base = b * N_;
    int g = sidx[b * S_ + s];
    const float* q = xyz + (size_t)(base + g) * 3;
    float qx = q[0], qy = q[1], qz = q[2];
    float qsq = sq[base + g];

    float bd0 = 3.4e38f, bd1 = 3.4e38f, bd2 = 3.4e38f, bd3 = 3.4e38f;
    int   bi0 = 0, bi1 = 0, bi2 = 0, bi3 = 0;

    constexpr int NT = N_ / 256;
    // prologue: async-stage tile 0 into buffer 0
    {
        const float* c = xyz + (size_t)(base + tid) * 3;
        async_dword(&cx[0][tid], c + 0);
        async_dword(&cy[0][tid], c + 1);
        async_dword(&cz[0][tid], c + 2);
        async_dword(&cs[0][tid], sq + base + tid);
    }
    for (int t = 0; t < NT; ++t) {
        int pb = t & 1;
        if (t + 1 < NT) {   // prefetch next tile into the other buffer
            int j = (t + 1) * 256 + tid;
            const float* c = xyz + (size_t)(base + j) * 3;
            async_dword(&cx[pb ^ 1][tid], c + 0);
            async_dword(&cy[pb ^ 1][tid], c + 1);
            async_dword(&cz[pb ^ 1][tid], c + 2);
            async_dword(&cs[pb ^ 1][tid], sq + base + j);
            asm volatile("s_wait_asynccnt 0x4" ::: "memory");  // current tile's 4 done (in-order)
        } else {
            asm volatile("s_wait_asynccnt 0x0" ::: "memory");
        }
        __syncthreads();
        for (int u = 0; u < 256; ++u) {
            float dot = qx * cx[pb][u] + qy * cy[pb][u] + qz * cz[pb][u];
            float d = sqrtf(fmaxf(qsq + cs[pb][u] - 2.0f * dot, 1e-12f));
            ins4(d, t * 256 + u, bd0, bd1, bd2, bd3, bi0, bi1, bi2, bi3);
        }
        __syncthreads();
    }

    float d12 = pdist(xyz, sq, base, bi1, bi2);
    float d13 = pdist(xyz, sq, base, bi1, bi3);
    float d23 = pdist(xyz, sq, base, bi2, bi3);

    float* tr = trans + (size_t)(b * S_ + s) * GEN_CH;
    tr[0] = bd1; tr[1] = bd2; tr[2] = bd3;
    tr[3] = d12; tr[4] = d13; tr[5] = d23;
    int* an = anchors + (size_t)(b * S_ + s) * 4;
    an[0] = bi0; an[1] = bi1; an[2] = bi2; an[3] = bi3;
}

// ---------------- KNN among sampled points + intra_after + inter ----------------
__global__ __launch_bounds__(256) void knn2_kernel(const float* __restrict__ xyz,
                                                   const float* __restrict__ sq,
                                                   const int* __restrict__ sidx,
                                                   const int* __restrict__ anchors,
                                                   float* __restrict__ trans) {
    __shared__ float cx[2][256], cy[2][256], cz[2][256], cs[2][256];
    __shared__ int   cg[2][256];
    int tid = threadIdx.x;
    int b = blockIdx.y;
    int s = blockIdx.x * 256 + tid;
    int base = b * N_;
    int g = sidx[b * S_ + s];
    const float* q = xyz + (size_t)(base + g) * 3;
    float qx = q[0], qy = q[1], qz = q[2];
    float qsq = sq[base + g];

    float bd0 = 3.4e38f, bd1 = 3.4e38f, bd2 = 3.4e38f, bd3 = 3.4e38f;
    int   bi0 = 0, bi1 = 0, bi2 = 0, bi3 = 0;   // GLOBAL ids (anchors_after)

    constexpr int NT = S_ / 256;
    {
        int jg = sidx[b * S_ + tid];
        cg[0][tid] = jg;
        const float* c = xyz + (size_t)(base + jg) * 3;
        async_dword(&cx[0][tid], c + 0);
        async_dword(&cy[0][tid], c + 1);
        async_dword(&cz[0][tid], c + 2);
        async_dword(&cs[0][tid], sq + base + jg);
    }
    for (int t = 0; t < NT; ++t) {
        int pb = t & 1;
        if (t + 1 < NT) {
            int jg = sidx[b * S_ + (t + 1) * 256 + tid];
            cg[pb ^ 1][tid] = jg;
            const float* c = xyz + (size_t)(base + jg) * 3;
            async_dword(&cx[pb ^ 1][tid], c + 0);
            async_dword(&cy[pb ^ 1][tid], c + 1);
            async_dword(&cz[pb ^ 1][tid], c + 2);
            async_dword(&cs[pb ^ 1][tid], sq + base + jg);
            asm volatile("s_wait_asynccnt 0x4" ::: "memory");
        } else {
            asm volatile("s_wait_asynccnt 0x0" ::: "memory");
        }
        __syncthreads();
        for (int u = 0; u < 256; ++u) {
            float dot = qx * cx[pb][u] + qy * cy[pb][u] + qz * cz[pb][u];
            float d = sqrtf(fmaxf(qsq + cs[pb][u] - 2.0f * dot, 1e-12f));
            ins4(d, cg[pb][u], bd0, bd1, bd2, bd3, bi0, bi1, bi2, bi3);
        }
        __syncthreads();
    }

    float* tr = trans + (size_t)(b * S_ + s) * GEN_CH;
    // intra_after
    tr[6]  = bd1; tr[7] = bd2; tr[8] = bd3;
    tr[9]  = pdist(xyz, sq, base, bi1, bi2);
    tr[10] = pdist(xyz, sq, base, bi1, bi3);
    tr[11] = pdist(xyz, sq, base, bi2, bi3);
    // inter: 4x4 between anchors_before (i) and anchors_after (j)
    int ab[4], af[4];
    const int* an = anchors + (size_t)(b * S_ + s) * 4;
    ab[0] = an[0]; ab[1] = an[1]; ab[2] = an[2]; ab[3] = an[3];
    af[0] = bi0; af[1] = bi1; af[2] = bi2; af[3] = bi3;
#pragma unroll
    for (int i = 0; i < 4; ++i)
#pragma unroll
        for (int j = 0; j < 4; ++j)
            tr[12 + i * 4 + j] = pdist(xyz, sq, base, ab[i], af[j]);
}

// ---------------- fused 3-layer MLP via V_WMMA_F32_16X16X4_F32 ----------------
// Each wave owns a 32-row tile (two 16-row A sub-tiles sharing every B fragment),
// doubling WMMA issued per weight dword loaded.
#define WPB 2
__global__ __launch_bounds__(32 * WPB) void mlp_kernel(
    const float* __restrict__ trans, const float* __restrict__ feat,
    const int* __restrict__ sidx,
    const float* __restrict__ w1, const float* __restrict__ b1, const float* __restrict__ g1, const float* __restrict__ be1,
    const float* __restrict__ w2, const float* __restrict__ b2, const float* __restrict__ g2, const float* __restrict__ be2,
    const float* __restrict__ w3, const float* __restrict__ b3, const float* __restrict__ g3, const float* __restrict__ be3,
    float* __restrict__ outp) {
    __shared__ float h1[WPB][32 * 64];
    __shared__ float h2[WPB][32 * 64];
    const float RS = 0.99999500003749969f;   // 1/sqrt(1 + 1e-5)

    int wave = threadIdx.x >> 5;
    int l = threadIdx.x & 31;
    int half = l >> 4;
    int lane16 = l & 15;
    int koff = half * 2;
    int rowBase = (blockIdx.x * WPB + wave) * 32;
    const float* trow0 = trans + (size_t)(rowBase + lane16) * GEN_CH;
    const float* trow1 = trans + (size_t)(rowBase + 16 + lane16) * GEN_CH;

    // ---- layer 1: 28 -> 64 ----
#pragma unroll
    for (int ct = 0; ct < 4; ++ct) {
        int col = ct * 16 + lane16;
        v8f acc0 = {0.f,0.f,0.f,0.f,0.f,0.f,0.f,0.f};
        v8f acc1 = {0.f,0.f,0.f,0.f,0.f,0.f,0.f,0.f};
#pragma unroll
        for (int ks = 0; ks < GEN_CH; ks += 4) {
            int k0 = ks + koff;
            v2f bb; bb.x = w1[k0 * 64 + col]; bb.y = w1[(k0 + 1) * 64 + col];
            v2f a0; a0.x = trow0[k0]; a0.y = trow0[k0 + 1];
            v2f a1; a1.x = trow1[k0]; a1.y = trow1[k0 + 1];
            acc0 = wmma_f32_k4(a0, bb, acc0);
            acc1 = wmma_f32_k4(a1, bb, acc1);
        }
        float sc = g1[col] * RS, bo = b1[col], bt = be1[col];
#pragma unroll
        for (int r = 0; r < 8; ++r) {
            int m = r + half * 8;
            float v0 = (acc0[r] + bo) * sc + bt;
            v0 = v0 >= 0.f ? v0 : 0.2f * v0;
            h1[wave][m * 64 + col] = v0;
            float v1 = (acc1[r] + bo) * sc + bt;
            v1 = v1 >= 0.f ? v1 : 0.2f * v1;
            h1[wave][(16 + m) * 64 + col] = v1;
        }
    }
    __syncthreads();

    // ---- layer 2: 64 -> 64 ----
#pragma unroll
    for (int ct = 0; ct < 4; ++ct) {
        int col = ct * 16 + lane16;
        v8f acc0 = {0.f,0.f,0.f,0.f,0.f,0.f,0.f,0.f};
        v8f acc1 = {0.f,0.f,0.f,0.f,0.f,0.f,0.f,0.f};
#pragma unroll
        for (int ks = 0; ks < 64; ks += 4) {
            int k0 = ks + koff;
            v2f bb; bb.x = w2[k0 * 64 + col]; bb.y = w2[(k0 + 1) * 64 + col];
            v2f a0; a0.x = h1[wave][lane16 * 64 + k0];
                    a0.y = h1[wave][lane16 * 64 + k0 + 1];
            v2f a1; a1.x = h1[wave][(16 + lane16) * 64 + k0];
                    a1.y = h1[wave][(16 + lane16) * 64 + k0 + 1];
            acc0 = wmma_f32_k4(a0, bb, acc0);
            acc1 = wmma_f32_k4(a1, bb, acc1);
        }
        float sc = g2[col] * RS, bo = b2[col], bt = be2[col];
#pragma unroll
        for (int r = 0; r < 8; ++r) {
            int m = r + half * 8;
            float v0 = (acc0[r] + bo) * sc + bt;
            v0 = v0 >= 0.f ? v0 : 0.2f * v0;
            h2[wave][m * 64 + col] = v0;
            float v1 = (acc1[r] + bo) * sc + bt;
            v1 = v1 >= 0.f ? v1 : 0.2f * v1;
            h2[wave][(16 + m) * 64 + col] = v1;
        }
    }
    __syncthreads();

    // ---- stage gathered feature rows into (now free) h1 ----
    for (int i = l; i < 32 * 64; i += 32) {
        int m = i >> 6, c = i & 63;
        int row = rowBase + m;                 // b*S + s
        int bb = row >> 12;                    // / S_
        int gg = sidx[row];
        h1[wave][i] = feat[((size_t)bb * N_ + gg) * IN_CH + c];
    }
    __syncthreads();

    // ---- layer 3: concat(feature 64, h2 64) = 128 -> 128 ----
#pragma unroll
    for (int ct = 0; ct < 8; ++ct) {
        int col = ct * 16 + lane16;
        v8f acc0 = {0.f,0.f,0.f,0.f,0.f,0.f,0.f,0.f};
        v8f acc1 = {0.f,0.f,0.f,0.f,0.f,0.f,0.f,0.f};
#pragma unroll
        for (int ks = 0; ks < 128; ks += 4) {
            int k0 = ks + koff;                // k0,k0+1 stay on the same side of 64
            v2f bb; bb.x = w3[k0 * 128 + col]; bb.y = w3[(k0 + 1) * 128 + col];
            v2f a0, a1;
            if (k0 < 64) {
                a0.x = h1[wave][lane16 * 64 + k0];
                a0.y = h1[wave][lane16 * 64 + k0 + 1];
                a1.x = h1[wave][(16 + lane16) * 64 + k0];
                a1.y = h1[wave][(16 + lane16) * 64 + k0 + 1];
            } else {
                a0.x = h2[wave][lane16 * 64 + k0 - 64];
                a0.y = h2[wave][lane16 * 64 + k0 - 63];
                a1.x = h2[wave][(16 + lane16) * 64 + k0 - 64];
                a1.y = h2[wave][(16 + lane16) * 64 + k0 - 63];
            }
            acc0 = wmma_f32_k4(a0, bb, acc0);
            acc1 = wmma_f32_k4(a1, bb, acc1);
        }
        float sc = g3[col] * RS, bo = b3[col], bt = be3[col];
#pragma unroll
        for (int r = 0; r < 8; ++r) {
            int m = r + half * 8;
            float v0 = (acc0[r] + bo) * sc + bt;
            v0 = v0 >= 0.f ? v0 : 0.2f * v0;
            outp[(size_t)(rowBase + m) * OUT_CH + col] = v0;
            float v1 = (acc1[r] + bo) * sc + bt;
            v1 = v1 >= 0.f ? v1 : 0.2f * v1;
            outp[(size_t)(rowBase + 16 + m) * OUT_CH + col] = v1;
        }
    }
}

// ---------------- launch ----------------
extern "C" void kernel_launch(void* const* d_in, const int* in_sizes, int n_in,
                              void* d_out, int out_size, void* d_ws, size_t ws_size,
                              hipStream_t stream) {
    (void)in_sizes; (void)n_in; (void)out_size; (void)ws_size;
    const float* xyz  = (const float*)d_in[0];
    const float* feat = (const float*)d_in[1];
    const int*   sidx = (const int*)  d_in[2];
    const float* w1  = (const float*)d_in[3];
    const float* b1  = (const float*)d_in[4];
    const float* g1  = (const float*)d_in[5];
    const float* be1 = (const float*)d_in[6];
    const float* w2  = (const float*)d_in[7];
    const float* b2  = (const float*)d_in[8];
    const float* g2  = (const float*)d_in[9];
    const float* be2 = (const float*)d_in[10];
    const float* w3  = (const float*)d_in[11];
    const float* b3  = (const float*)d_in[12];
    const float* g3  = (const float*)d_in[13];
    const float* be3 = (const float*)d_in[14];
    float* out = (float*)d_out;

    // workspace layout
    size_t sq_bytes   = (size_t)B_ * N_ * sizeof(float);           // 64 KB
    size_t anch_bytes = (size_t)B_ * S_ * 4 * sizeof(int);         // 128 KB
    float* sq      = (float*)d_ws;
    int*   anchors = (int*)((char*)d_ws + sq_bytes);
    float* trans   = (float*)((char*)d_ws + sq_bytes + anch_bytes); // B*S*28 f32

    sq_kernel<<<(B_ * N_ + 255) / 256, 256, 0, stream>>>(xyz, sq);
    sampled_xyz_kernel<<<(B_ * S_ + 255) / 256, 256, 0, stream>>>(xyz, sidx, out);

    dim3 kgrid(S_ / 256, B_);
    knn1_kernel<<<kgrid, 256, 0, stream>>>(xyz, sq, sidx, anchors, trans);
    knn2_kernel<<<kgrid, 256, 0, stream>>>(xyz, sq, sidx, anchors, trans);

    int rowsPerBlock = WPB * 32;            // 64 rows per block
    mlp_kernel<<<(B_ * S_) / rowsPerBlock, 32 * WPB, 0, stream>>>(
        trans, feat, sidx,
        w1, b1, g1, be1, w2, b2, g2, be2, w3, b3, g3, be3,
        out + (size_t)B_ * S_ * 3);
}